// HungarianMatcher_7335804141876
// MI455X (gfx1250) — compile-verified
//
#include <hip/hip_runtime.h>
#include <hip/hip_bf16.h>
#include <math.h>

#define B_  16
#define Q_  600
#define C_  91
#define CP  96            // C padded to multiple of 4 (WMMA K-chunks), padded cols are 0
#define T_  1024
#define BQ  (B_ * Q_)     // 9600

typedef float v2f __attribute__((ext_vector_type(2)));
typedef float v8f __attribute__((ext_vector_type(8)));

// ---------------------------------------------------------------------------
// Pass 1: row softmax over C=91, written padded to CP=96 (pad cols = 0.0f).
// One wave32 per row; 8 waves per 256-thread block.
// ---------------------------------------------------------------------------
__global__ __launch_bounds__(256) void hm_softmax_kernel(
    const float* __restrict__ logits, float* __restrict__ P) {
  const int wave = threadIdx.x >> 5;
  const int lane = threadIdx.x & 31;
  const int row  = blockIdx.x * 8 + wave;
  if (row >= BQ) return;

  const float* x = logits + row * C_;
  float x0 = x[lane];                                   // cols 0..31
  float x1 = x[lane + 32];                              // cols 32..63
  float x2 = (lane + 64 < C_) ? x[lane + 64] : -INFINITY; // cols 64..90

  float m = fmaxf(fmaxf(x0, x1), x2);
  #pragma unroll
  for (int off = 16; off >= 1; off >>= 1)
    m = fmaxf(m, __shfl_xor(m, off, 32));

  float e0 = __expf(x0 - m);
  float e1 = __expf(x1 - m);
  float e2 = (lane + 64 < C_) ? __expf(x2 - m) : 0.0f;

  float s = e0 + e1 + e2;
  #pragma unroll
  for (int off = 16; off >= 1; off >>= 1)
    s += __shfl_xor(s, off, 32);

  const float inv = 1.0f / s;
  float* p = P + row * CP;
  p[lane]      = e0 * inv;
  p[lane + 32] = e1 * inv;
  p[lane + 64] = e2 * inv;   // cols 91..95 get 0.0f (e2 forced to 0)
}

// ---------------------------------------------------------------------------
// Pass 2: fused cost matrix. One wave per (16-row block, 128-target group).
// cost_class via V_WMMA_F32_16X16X4_F32 against an on-the-fly one-hot B.
// ---------------------------------------------------------------------------
__global__ __launch_bounds__(32) void hm_cost_kernel(
    const float* __restrict__ P,            // [BQ, CP] softmax, zero-padded
    const float* __restrict__ pred_boxes,   // [BQ, 4] cxcywh
    const int*   __restrict__ tgt_labels,   // [T]
    const float* __restrict__ tgt_boxes,    // [T, 4] cxcywh
    float*       __restrict__ out) {        // [BQ, T]
  const int lane   = threadIdx.x;          // 0..31, whole wave active
  const int mlo    = lane & 15;
  const int khalf  = lane >> 4;            // 0: K pair {0,1}, 1: K pair {2,3}
  const int row_base  = blockIdx.x * 16;   // 600 row blocks
  const int col_group = blockIdx.y * 128;  // 8 column groups

  // ---- A fragments for this 16-row block: reused across all 8 N-tiles ----
  // A(16x4 f32) layout: lane holds P[row_base + mlo][4*kc + 2*khalf + {0,1}]
  v2f a[CP / 4];
  {
    const float* prow = P + (size_t)(row_base + mlo) * CP + 2 * khalf;
    #pragma unroll
    for (int kc = 0; kc < CP / 4; ++kc)
      a[kc] = *(const v2f*)(prow + 4 * kc);   // 8B-aligned
  }

  // ---- Pred boxes for this lane's 8 output rows: M = 8*khalf + r ----
  float pcx[8], pcy[8], pw[8], ph[8];
  float px0[8], py0[8], px1[8], py1[8], pa[8];
  #pragma unroll
  for (int r = 0; r < 8; ++r) {
    const float4 pb = *(const float4*)(pred_boxes + (size_t)(row_base + 8 * khalf + r) * 4);
    pcx[r] = pb.x; pcy[r] = pb.y; pw[r] = pb.z; ph[r] = pb.w;
    px0[r] = pb.x - 0.5f * pb.z;  py0[r] = pb.y - 0.5f * pb.w;
    px1[r] = pb.x + 0.5f * pb.z;  py1[r] = pb.y + 0.5f * pb.w;
    pa[r]  = pb.z * pb.w;
  }

  #pragma unroll 1
  for (int nt = 0; nt < 8; ++nt) {
    const int col = col_group + nt * 16 + mlo;   // this lane's target column
    const int lbl = tgt_labels[col];
    const float4 tb = *(const float4*)(tgt_boxes + (size_t)col * 4);
    const float tcx = tb.x, tcy = tb.y, tw = tb.z, th = tb.w;
    const float tx0 = tcx - 0.5f * tw, ty0 = tcy - 0.5f * th;
    const float tx1 = tcx + 0.5f * tw, ty1 = tcy + 0.5f * th;
    const float ta  = tw * th;

    // ---- cost_class accumulation: D = A x OneHot + D, exact in f32 ----
    v8f c = {};
    #pragma unroll
    for (int kc = 0; kc < CP / 4; ++kc) {
      const int kb = 4 * kc + 2 * khalf;
      v2f b;
      b.x = (lbl == kb)     ? 1.0f : 0.0f;
      b.y = (lbl == kb + 1) ? 1.0f : 0.0f;
      c = __builtin_amdgcn_wmma_f32_16x16x4_f32(
          /*neg_a=*/false, a[kc], /*neg_b=*/false, b,
          /*c_mod=*/(short)0, c, /*reuse_a=*/false, /*reuse_b=*/false);
    }

    // ---- combine with L1 + GIoU for this lane's 8 (M,N) pairs ----
    #pragma unroll
    for (int r = 0; r < 8; ++r) {
      const float l1 = fabsf(pcx[r] - tcx) + fabsf(pcy[r] - tcy) +
                       fabsf(pw[r]  - tw ) + fabsf(ph[r]  - th );
      const float ix0 = fmaxf(px0[r], tx0), iy0 = fmaxf(py0[r], ty0);
      const float ix1 = fminf(px1[r], tx1), iy1 = fminf(py1[r], ty1);
      const float iw = fmaxf(ix1 - ix0, 0.0f), ih = fmaxf(iy1 - iy0, 0.0f);
      const float inter = iw * ih;
      const float uni   = pa[r] + ta - inter;
      const float iou   = inter / uni;
      const float cx0 = fminf(px0[r], tx0), cy0 = fminf(py0[r], ty0);
      const float cx1 = fmaxf(px1[r], tx1), cy1 = fmaxf(py1[r], ty1);
      const float cw = fmaxf(cx1 - cx0, 0.0f), ch = fmaxf(cy1 - cy0, 0.0f);
      const float ac = cw * ch;
      const float giou = iou - (ac - uni) / ac;

      const int rowm = row_base + 8 * khalf + r;
      // cost = 5*L1 + 1*(-P[row,label]) + 2*(-giou);  c[r] holds +P
      out[(size_t)rowm * T_ + col] = 5.0f * l1 - 2.0f * giou - c[r];
    }
  }
}

extern "C" void kernel_launch(void* const* d_in, const int* in_sizes, int n_in,
                              void* d_out, int out_size, void* d_ws, size_t ws_size,
                              hipStream_t stream) {
  const float* logits = (const float*)d_in[0];   // [B,Q,C] f32
  const float* pboxes = (const float*)d_in[1];   // [B,Q,4] f32
  const int*   labels = (const int*)  d_in[2];   // [T] int
  const float* tboxes = (const float*)d_in[3];   // [T,4] f32
  float* out = (float*)d_out;                    // [B,Q,T] f32
  float* P   = (float*)d_ws;                     // [BQ, CP] f32 = 3.52 MB scratch

  hm_softmax_kernel<<<BQ / 8, 256, 0, stream>>>(logits, P);

  dim3 grid(BQ / 16, T_ / 128);                  // 600 x 8 waves
  hm_cost_kernel<<<grid, 32, 0, stream>>>(P, pboxes, labels, tboxes, out);
}